// MyCustomLightGCN_72971494359151
// MI455X (gfx1250) — compile-verified
//
#include <hip/hip_runtime.h>

#define NNODES  50000
#define NEDGES  800000
#define DIM     64
#define NGROUPS (NNODES / 16)   // 3125, exact

typedef __attribute__((ext_vector_type(2))) float v2f;
typedef __attribute__((ext_vector_type(8))) float v8f;

// ---------------- helpers ----------------

__global__ void k_zero2(int* __restrict__ a, int* __restrict__ b, int n) {
    int i = blockIdx.x * blockDim.x + threadIdx.x;
    if (i < n) { a[i] = 0; b[i] = 0; }
}

__global__ void k_deg(const int* __restrict__ dst, int* __restrict__ deg, int E) {
    int e = blockIdx.x * blockDim.x + threadIdx.x;
    if (e < E) atomicAdd(&deg[dst[e]], 1);
}

__global__ void k_dinv(const int* __restrict__ deg, float* __restrict__ dinv, int n) {
    int i = blockIdx.x * blockDim.x + threadIdx.x;
    if (i < n) {
        int d = deg[i];
        dinv[i] = (d > 0) ? rsqrtf((float)d) : 0.0f;
    }
}

// single-block exclusive scan: rowstart[0..n], rowstart[n] = total
__global__ void k_scan(const int* __restrict__ deg, int* __restrict__ rowstart, int n) {
    __shared__ int tmp[1024];
    __shared__ int carry;
    if (threadIdx.x == 0) carry = 0;
    __syncthreads();
    for (int base = 0; base < n; base += 1024) {
        int i = base + (int)threadIdx.x;
        int v = (i < n) ? deg[i] : 0;
        tmp[threadIdx.x] = v;
        __syncthreads();
        for (int off = 1; off < 1024; off <<= 1) {
            int t = (threadIdx.x >= (unsigned)off) ? tmp[threadIdx.x - off] : 0;
            __syncthreads();
            tmp[threadIdx.x] += t;
            __syncthreads();
        }
        int incl = tmp[threadIdx.x];
        if (i < n) rowstart[i] = carry + (incl - v);
        __syncthreads();
        if (threadIdx.x == 1023) carry += incl;  // incl here == block total
        __syncthreads();
    }
    if (threadIdx.x == 0) rowstart[n] = carry;
}

// scatter edges into dst-sorted CSR; one 64-bit record per edge:
//   low 32:  src (20 bits) | localrow (bits 20..23)
//   high 32: bit-cast f32 weight
__global__ void k_fill(const int* __restrict__ src, const int* __restrict__ dst,
                       const float* __restrict__ dinv, const int* __restrict__ rowstart,
                       int* __restrict__ cursor, unsigned long long* __restrict__ csr,
                       int E) {
    int e = blockIdx.x * blockDim.x + threadIdx.x;
    if (e >= E) return;
    int s = src[e];
    int d = dst[e];
    int pos = rowstart[d] + atomicAdd(&cursor[d], 1);
    unsigned lo = (unsigned)(s | ((d & 15) << 20));
    unsigned hi = (unsigned)__float_as_int(dinv[s] * dinv[d]);
    csr[pos] = ((unsigned long long)hi << 32) | lo;
}

__global__ void k_copy(const float* __restrict__ a, float* __restrict__ o, int n) {
    int i = blockIdx.x * blockDim.x + threadIdx.x;
    if (i < n) o[i] = a[i];
}

__global__ void k_scale(float* __restrict__ o, float s, int n) {
    int i = blockIdx.x * blockDim.x + threadIdx.x;
    if (i < n) o[i] *= s;
}

// ---------------- WMMA SpMM ----------------
// One block = one 16-dst-node group; 4 waves = 4 slices of 16 dims.
// Per 4-edge chunk:  C(16x16) += A(16x4 selection*weight) x B(4x16 gathered src rows)
// via V_WMMA_F32_16X16X4_F32 (full fp32 path). Group owns its 16 output rows
// exclusively -> plain stores, no atomics. Main loop is branchless; A entries
// are produced with a bit-mask (hi & -(r==m)) so the weight word is used
// unconditionally and the compiler cannot predicate any load.
__global__ void __launch_bounds__(128)
k_spmm(const float* __restrict__ xin, float* __restrict__ xout,
       float* __restrict__ acc,
       const int* __restrict__ rowstart,
       const unsigned long long* __restrict__ csr) {
    const int g     = blockIdx.x;
    const int slice = threadIdx.x >> 5;   // 0..3 (wave id)
    const int lane  = threadIdx.x & 31;
    const int half  = lane >> 4;          // 0: K=0,1  1: K=2,3
    const int m     = lane & 15;          // A row / B col / D col
    const int nb    = slice * 16;
    const int node0 = g * 16;

    const int beg   = rowstart[node0];
    const int end   = rowstart[node0 + 16];
    const int nfull = (end - beg) & ~3;
    const int fend  = beg + nfull;

    v8f c = {};
    int p = beg;
#pragma unroll 2
    for (; p < fend; p += 4) {
        const int e0 = p + 2 * half;      // this half's first K slot
        unsigned long long q0 = csr[e0];
        unsigned long long q1 = csr[e0 + 1];
        int x0 = (int)(unsigned)q0;
        int x1 = (int)(unsigned)q1;
        int s0 = x0 & 0xFFFFF, r0 = x0 >> 20;
        int s1 = x1 & 0xFFFFF, r1 = x1 >> 20;
        float b0 = xin[s0 * DIM + nb + m];
        float b1 = xin[s1 * DIM + nb + m];
        // branchless: weight word ANDed with all-ones/all-zeros mask
        float a0 = __int_as_float((int)(q0 >> 32) & -(int)(r0 == m));
        float a1 = __int_as_float((int)(q1 >> 32) & -(int)(r1 == m));
        v2f A; A.x = a0; A.y = a1;        // A[m, 2*half], A[m, 2*half+1]
        v2f B; B.x = b0; B.y = b1;        // B[2*half, m], B[2*half+1, m]
        c = __builtin_amdgcn_wmma_f32_16x16x4_f32(
                false, A, false, B, (short)0, c, false, false);
    }

    if (p < end) {                        // wave-uniform: 1..3 leftover edges
        const int e0 = p + 2 * half;
        float a0 = 0.0f, a1 = 0.0f, b0 = 0.0f, b1 = 0.0f;
        if (e0 < end) {
            unsigned long long q0 = csr[e0];
            int x0 = (int)(unsigned)q0;
            int s0 = x0 & 0xFFFFF, r0 = x0 >> 20;
            a0 = __int_as_float((int)(q0 >> 32) & -(int)(r0 == m));
            b0 = xin[s0 * DIM + nb + m];
        }
        if (e0 + 1 < end) {
            unsigned long long q1 = csr[e0 + 1];
            int x1 = (int)(unsigned)q1;
            int s1 = x1 & 0xFFFFF, r1 = x1 >> 20;
            a1 = __int_as_float((int)(q1 >> 32) & -(int)(r1 == m));
            b1 = xin[s1 * DIM + nb + m];
        }
        v2f A; A.x = a0; A.y = a1;
        v2f B; B.x = b0; B.y = b1;
        c = __builtin_amdgcn_wmma_f32_16x16x4_f32(
                false, A, false, B, (short)0, c, false, false);
    }

    // D layout: vgpr j holds row (j + 8*half), col = m
#pragma unroll
    for (int j = 0; j < 8; ++j) {
        int   node = node0 + half * 8 + j;
        int   idx  = node * DIM + nb + m;
        float v    = c[j];
        xout[idx] = v;      // next-layer input (every element written, no pre-zero)
        acc[idx] += v;      // running sum for final average (exclusive rows)
    }
}

// ---------------- launch ----------------

extern "C" void kernel_launch(void* const* d_in, const int* in_sizes, int n_in,
                              void* d_out, int out_size, void* d_ws, size_t ws_size,
                              hipStream_t stream) {
    const int*   edge = (const int*)d_in[0];       // (2, E) int32
    const int*   src  = edge;
    const int*   dst  = edge + NEDGES;
    const float* emb  = (const float*)d_in[1];     // (N, 64) f32
    float*       out  = (float*)d_out;

    char* ws = (char*)d_ws;
    int*   deg      = (int*)  (ws + 0);                     // 200 KB
    int*   cursor   = (int*)  (ws + (256u << 10));          // 200 KB
    float* dinv     = (float*)(ws + (512u << 10));          // 200 KB
    int*   rowstart = (int*)  (ws + (768u << 10));          // 200 KB (+1)
    unsigned long long* csr = (unsigned long long*)(ws + (1u << 20)); // 6.4 MB
    float* bufA     = (float*)(ws + (8u   << 20));          // 12.8 MB
    float* bufB     = (float*)(ws + (21u  << 20));          // 12.8 MB

    const int TB = 256;
    k_zero2<<<(NNODES + TB - 1) / TB, TB, 0, stream>>>(deg, cursor, NNODES);
    k_deg  <<<(NEDGES + TB - 1) / TB, TB, 0, stream>>>(dst, deg, NEDGES);
    k_dinv <<<(NNODES + TB - 1) / TB, TB, 0, stream>>>(deg, dinv, NNODES);
    k_scan <<<1, 1024, 0, stream>>>(deg, rowstart, NNODES);
    k_fill <<<(NEDGES + TB - 1) / TB, TB, 0, stream>>>(src, dst, dinv, rowstart,
                                                       cursor, csr, NEDGES);

    const int NE = NNODES * DIM;
    k_copy<<<(NE + TB - 1) / TB, TB, 0, stream>>>(emb, out, NE);   // acc = x0

    k_spmm<<<NGROUPS, 128, 0, stream>>>(emb,  bufA, out, rowstart, csr); // x1
    k_spmm<<<NGROUPS, 128, 0, stream>>>(bufA, bufB, out, rowstart, csr); // x2
    k_spmm<<<NGROUPS, 128, 0, stream>>>(bufB, bufA, out, rowstart, csr); // x3

    k_scale<<<(NE + TB - 1) / TB, TB, 0, stream>>>(out, 0.25f, NE);
}